// OrderLossFromFeatures_412316860463
// MI455X (gfx1250) — compile-verified
//
#include <hip/hip_runtime.h>
#include <hip/hip_bf16.h>

typedef float v2f __attribute__((ext_vector_type(2)));
typedef float v8f __attribute__((ext_vector_type(8)));

// Exact wave32 sum using V_WMMA_F32_16X16X4_F32 (full fp32, no precision loss).
// A (16x4): VGPR0 = per-lane value -> A[m][0]=v[m] (lanes 0-15), A[m][2]=v[m+16]
//           VGPR1 = 0             -> A[m][1]=A[m][3]=0
// B (4x16): all ones  =>  D[m][n] = v[m] + v[m+16] for every n.
// D VGPR g: lanes 0-15 = v[g]+v[g+16], lanes 16-31 = v[g+8]+v[g+24].
// Sum the 8 D VGPRs, fold halves with one shuffle -> full wave sum in all lanes.
// Must be called at full EXEC (all 32 lanes active).
__device__ __forceinline__ float wave_sum_wmma(float v) {
    v2f a; a[0] = v;     a[1] = 0.0f;
    v2f b; b[0] = 1.0f;  b[1] = 1.0f;
    v8f c = {};
    c = __builtin_amdgcn_wmma_f32_16x16x4_f32(
        /*neg_a=*/false, a, /*neg_b=*/false, b,
        /*c_mod=*/(short)0, c, /*reuse_a=*/false, /*reuse_b=*/false);
    float s = ((c[0] + c[1]) + (c[2] + c[3])) + ((c[4] + c[5]) + (c[6] + c[7]));
    s += __shfl_xor(s, 16, 32);
    return s;
}

__device__ __forceinline__ float softplus_stable(float x) {
    // logaddexp(0, x) = max(x,0) + log1p(exp(-|x|))
    return fmaxf(x, 0.0f) + log1pf(expf(-fabsf(x)));
}

// Block-level finish: wave WMMA-reduce two accumulators, cross-wave via LDS,
// thread 0 writes (sum, count) to partials[blockIdx.x].
__device__ __forceinline__ void block_finish2(float lsum, float lcnt,
                                              float* __restrict__ partials) {
    float ws = wave_sum_wmma(lsum);
    float wc = wave_sum_wmma(lcnt);
    __shared__ float lds[8][2];
    int wave = threadIdx.x >> 5, lane = threadIdx.x & 31;
    if (lane == 0) { lds[wave][0] = ws; lds[wave][1] = wc; }
    __syncthreads();
    if (threadIdx.x == 0) {
        float bs = 0.0f, bc = 0.0f;
        for (int w = 0; w < 8; ++w) { bs += lds[w][0]; bc += lds[w][1]; }
        partials[blockIdx.x * 2 + 0] = bs;
        partials[blockIdx.x * 2 + 1] = bc;
    }
}

// ---------------- Kernel A (vectorized, requires N % 4 == 0) ----------------
// Streams order_logits as float4 (b128); a 4-chunk never crosses a row, so
// (b,i) is constant per chunk: i-side mask/order loaded once, j-side as u32/int4.
__global__ __launch_bounds__(256) void order_partial_vec_kernel(
    const float* __restrict__ order_logits,
    const int* __restrict__ reading_orders,
    const unsigned char* __restrict__ region_mask,
    float* __restrict__ partials,   // [gridDim.x][2]
    int B, int N) {
    const int NN = N * N;
    const long long M = (long long)B * NN;
    const long long C = M >> 2;                         // float4 chunks
    const long long stride = (long long)gridDim.x * blockDim.x;

    float lsum = 0.0f, lcnt = 0.0f;
    for (long long c4 = (long long)blockIdx.x * blockDim.x + threadIdx.x;
         c4 < C; c4 += stride) {
        long long base = c4 << 2;
        int b   = (int)(base / NN);
        int rem = (int)(base - (long long)b * NN);
        int i   = rem / N;
        int j0  = rem - i * N;                          // multiple of 4
        const int  rowOff = b * N;

        float4 x = *reinterpret_cast<const float4*>(order_logits + base);          // b128
        int4  roj = *reinterpret_cast<const int4*>(reading_orders + rowOff + j0);  // b128
        unsigned int mjw =
            *reinterpret_cast<const unsigned int*>(region_mask + rowOff + j0);     // 4 bytes
        bool mi  = region_mask[rowOff + i] != 0;
        int  roi = reading_orders[rowOff + i];

        float xs[4] = {x.x, x.y, x.z, x.w};
        int   rj[4] = {roj.x, roj.y, roj.z, roj.w};
#pragma unroll
        for (int k = 0; k < 4; ++k) {
            int j = j0 + k;
            bool mask = mi && ((mjw >> (8 * k)) & 0xffu) && (i != j);
            if (mask) {
                float t = (roi < rj[k]) ? 1.0f : 0.0f;
                lsum += softplus_stable(xs[k]) - xs[k] * t;
                lcnt += 1.0f;
            }
        }
    }
    // scalar tail (empty when M % 4 == 0)
    for (long long idx = (C << 2) + (long long)blockIdx.x * blockDim.x + threadIdx.x;
         idx < M; idx += stride) {
        int b   = (int)(idx / NN);
        int rem = (int)(idx - (long long)b * NN);
        int i   = rem / N;
        int j   = rem - i * N;
        float x = order_logits[idx];
        bool mask = (i != j) && region_mask[b * N + i] && region_mask[b * N + j];
        if (mask) {
            float t = (reading_orders[b * N + i] < reading_orders[b * N + j]) ? 1.0f : 0.0f;
            lsum += softplus_stable(x) - x * t;
            lcnt += 1.0f;
        }
    }
    block_finish2(lsum, lcnt, partials);
}

// ---------------- Kernel A (scalar fallback for N % 4 != 0) ----------------
__global__ __launch_bounds__(256) void order_partial_kernel(
    const float* __restrict__ order_logits,
    const int* __restrict__ reading_orders,
    const unsigned char* __restrict__ region_mask,
    float* __restrict__ partials,
    int B, int N) {
    const int NN = N * N;
    const long long M = (long long)B * NN;
    const long long stride = (long long)gridDim.x * blockDim.x;

    float lsum = 0.0f, lcnt = 0.0f;
    for (long long idx = (long long)blockIdx.x * blockDim.x + threadIdx.x;
         idx < M; idx += stride) {
        int b   = (int)(idx / NN);
        int rem = (int)(idx - (long long)b * NN);
        int i   = rem / N;
        int j   = rem - i * N;
        float x = order_logits[idx];
        bool mask = (i != j) && region_mask[b * N + i] && region_mask[b * N + j];
        if (mask) {
            float t = (reading_orders[b * N + i] < reading_orders[b * N + j]) ? 1.0f : 0.0f;
            lsum += softplus_stable(x) - x * t;
            lcnt += 1.0f;
        }
    }
    block_finish2(lsum, lcnt, partials);
}

// ---------------- Kernel B: gathered relation CE (one thread per (b,i) row) ----------------
__global__ __launch_bounds__(256) void relation_partial_kernel(
    const float* __restrict__ relation_logits,
    const int* __restrict__ parent_ids,
    const int* __restrict__ relations,
    const unsigned char* __restrict__ region_mask,
    float* __restrict__ partials,
    int B, int N, int R) {
    const int rows = B * N;
    const int stride = gridDim.x * blockDim.x;

    float lsum = 0.0f, lcnt = 0.0f;
    for (int r = blockIdx.x * blockDim.x + threadIdx.x; r < rows; r += stride) {
        int b = r / N;
        int i = r - b * N;
        int parent = parent_ids[r];
        int rel    = relations[r];
        bool valid = region_mask[r] && (parent >= 0) && (parent < N) && (rel >= 0);
        int p  = min(max(parent, 0), N - 1);
        int rc = min(max(rel, 0), R - 1);
        const float* pl = relation_logits + (((long long)b * N + p) * N + i) * R;

        float m, se, sel;
        if ((R & 3) == 0) {                       // vector path (R==16 in reference)
            m = -3.402823466e+38f; sel = 0.0f;
            for (int k = 0; k < R; k += 4) {
                float4 q = *reinterpret_cast<const float4*>(pl + k);   // 16B-aligned
                m = fmaxf(fmaxf(fmaxf(m, q.x), fmaxf(q.y, q.z)), q.w);
            }
            se = 0.0f;
            for (int k = 0; k < R; k += 4) {
                float4 q = *reinterpret_cast<const float4*>(pl + k);
                se += expf(q.x - m) + expf(q.y - m) + expf(q.z - m) + expf(q.w - m);
                if (rc >= k && rc < k + 4) {
                    float vv[4] = {q.x, q.y, q.z, q.w};
                    sel = vv[rc - k];
                }
            }
        } else {                                  // generic scalar fallback
            m = pl[0];
            for (int k = 1; k < R; ++k) m = fmaxf(m, pl[k]);
            se = 0.0f;
            for (int k = 0; k < R; ++k) se += expf(pl[k] - m);
            sel = pl[rc];
        }
        float ce = (m + logf(se)) - sel;
        if (valid) { lsum += ce; lcnt += 1.0f; }
    }
    block_finish2(lsum, lcnt, partials);
}

// ---------------- Kernel C: deterministic finalize (single block) ----------------
__global__ __launch_bounds__(256) void finalize_kernel(
    const float* __restrict__ orderPartials, int nOrder,
    const float* __restrict__ relPartials,   int nRel,
    float* __restrict__ out) {
    float os = 0.0f, oc = 0.0f, rs = 0.0f, rcnt = 0.0f;
    for (int i = threadIdx.x; i < nOrder; i += blockDim.x) {
        os += orderPartials[2 * i + 0];
        oc += orderPartials[2 * i + 1];
    }
    for (int i = threadIdx.x; i < nRel; i += blockDim.x) {
        rs   += relPartials[2 * i + 0];
        rcnt += relPartials[2 * i + 1];
    }
    os = wave_sum_wmma(os); oc = wave_sum_wmma(oc);
    rs = wave_sum_wmma(rs); rcnt = wave_sum_wmma(rcnt);
    __shared__ float lds[8][4];
    int wave = threadIdx.x >> 5, lane = threadIdx.x & 31;
    if (lane == 0) {
        lds[wave][0] = os; lds[wave][1] = oc;
        lds[wave][2] = rs; lds[wave][3] = rcnt;
    }
    __syncthreads();
    if (threadIdx.x == 0) {
        float OS = 0, OC = 0, RS = 0, RC = 0;
        for (int w = 0; w < 8; ++w) {
            OS += lds[w][0]; OC += lds[w][1]; RS += lds[w][2]; RC += lds[w][3];
        }
        float order_loss = OS / fmaxf(OC, 1.0f);
        float rel_loss   = RS / fmaxf(RC, 1.0f);
        out[0] = 1.0f * order_loss + 0.5f * rel_loss;  // total
        out[1] = order_loss;
        out[2] = rel_loss;
    }
}

extern "C" void kernel_launch(void* const* d_in, const int* in_sizes, int n_in,
                              void* d_out, int out_size, void* d_ws, size_t ws_size,
                              hipStream_t stream) {
    const float*         order_logits    = (const float*)d_in[0];
    const float*         relation_logits = (const float*)d_in[1];
    const int*           reading_orders  = (const int*)d_in[2];
    const int*           parent_ids      = (const int*)d_in[3];
    const int*           relations       = (const int*)d_in[4];
    const unsigned char* region_mask     = (const unsigned char*)d_in[5]; // jnp.bool_ = 1 byte

    const int BN = in_sizes[2];                 // B*N
    const int N  = in_sizes[0] / BN;            // (B*N*N)/(B*N)
    const int B  = BN / N;
    const int R  = in_sizes[1] / in_sizes[0];   // (B*N*N*R)/(B*N*N)

    const int T = 256;
    const int GRID_A = 1024;                    // streaming blocks for order loss
    const int GRID_B = (BN + T - 1) / T;        // one thread per (b,i) row

    float* orderPart = (float*)d_ws;            // [GRID_A][2]
    float* relPart   = orderPart + GRID_A * 2;  // [GRID_B][2]

    if ((N & 3) == 0) {
        order_partial_vec_kernel<<<GRID_A, T, 0, stream>>>(
            order_logits, reading_orders, region_mask, orderPart, B, N);
    } else {
        order_partial_kernel<<<GRID_A, T, 0, stream>>>(
            order_logits, reading_orders, region_mask, orderPart, B, N);
    }
    relation_partial_kernel<<<GRID_B, T, 0, stream>>>(
        relation_logits, parent_ids, relations, region_mask, relPart, B, N, R);
    finalize_kernel<<<1, T, 0, stream>>>(
        orderPart, GRID_A, relPart, GRID_B, (float*)d_out);
}